// LNLinearAndKillingNonLinearAndPooling_94489280613
// MI455X (gfx1250) — compile-verified
//
#include <hip/hip_runtime.h>
#include <math.h>

typedef __attribute__((ext_vector_type(2))) float v2f;
typedef __attribute__((ext_vector_type(4))) float v4f;
typedef __attribute__((ext_vector_type(8))) float v8f;

#define BATCH  8
#define CIN    128
#define COUT   256
#define NDIM   4096
#define TILE_N 16
#define MCOLS  (8 * TILE_N)      // 128 flattened (k8,n) columns per tile
#define NT     (NDIM / TILE_N)   // 256 tiles per batch
#define PANEL  (COUT * MCOLS)    // 32768 floats per swizzled panel (128 KB)

// Fragment-native swizzled LDS layout for a [rows x 128] panel:
// element (row, col) lives at (row>>2)*512 + col*4 + (row&3)
// (row&3 == ((row>>1)&1)*2 + (row&1), i.e., WMMA-B pair order).
// Each lane's WMMA-B pair (K=k0+2p, K=k0+2p+1) is a contiguous 8B float2;
// rows 4c..4c+3 at one column are a contiguous 16B float4 (used for staging).
__device__ __forceinline__ int swz(int row, int col) {
    return ((row >> 2) << 9) + (col << 2) + (row & 3);
}

// Killing form with the fixed sparse sl(3) metric K6 = 6*G:
// G[0,2]=G[2,0]=G[1,4]=G[4,1]=G[3,5]=G[5,3]=1, G[6,6]=G[7,7]=2, G[6,7]=G[7,6]=-1
__device__ __forceinline__ float killing6(const float xv[8], const float dv[8]) {
    float s = xv[0]*dv[2] + xv[2]*dv[0]
            + xv[1]*dv[4] + xv[4]*dv[1]
            + xv[3]*dv[5] + xv[5]*dv[3]
            + 2.0f*(xv[6]*dv[6] + xv[7]*dv[7])
            - xv[6]*dv[7] - xv[7]*dv[6];
    return 6.0f * s;
}

// D[0:256, 0:128] = W[0:256, 0:K] x B[0:K, 0:128]; B and D swizzled in LDS.
// 8 waves; wave owns 32 output rows (2 row-tiles of 16), all 8 col-tiles.
__device__ __forceinline__ void gemm_block(const float* __restrict__ W, int ldw, int K,
                                           const float* __restrict__ Bsm,
                                           float* __restrict__ Dsm,
                                           int wave, int lane)
{
    const int l15 = lane & 15;
    const int p   = lane >> 4;   // half-wave: selects K-pair and C/D row stripe
    const int k2  = p << 1;
    const int rst = p << 3;

    for (int mi = wave * 2; mi < wave * 2 + 2; ++mi) {
        const int row0 = mi << 4;
        const float* Wrow = W + (size_t)(row0 + l15) * ldw;

        v8f acc[8];
        #pragma unroll
        for (int nj = 0; nj < 8; ++nj)
            acc[nj] = (v8f){0.f,0.f,0.f,0.f,0.f,0.f,0.f,0.f};

        // per-lane base into the swizzled B panel; nj adds constant 64-float steps
        const float* Bl = Bsm + (l15 << 2) + k2;

        #pragma unroll 2
        for (int kq4 = 0; kq4 < (K >> 2); ++kq4) {
            // A fragment (16x4 f32): lane<16 -> K=4kq,4kq+1 ; lane>=16 -> +2,+3
            v2f a = *(const v2f*)(Wrow + (kq4 << 2) + k2);
            const float* Bk = Bl + (kq4 << 9);
            #pragma unroll
            for (int nj = 0; nj < 8; ++nj) {
                v2f bb = *(const v2f*)(Bk + (nj << 6));     // ds_load_b64
                acc[nj] = __builtin_amdgcn_wmma_f32_16x16x4_f32(
                    false, a, false, bb, (short)0, acc[nj], false, false);
            }
        }
        // VGPR r holds D[row0 + r + 8p, col]; rows r,r+1 adjacent in swizzle.
        #pragma unroll
        for (int nj = 0; nj < 8; ++nj) {
            const int col = (nj << 4) + l15;
            #pragma unroll
            for (int r = 0; r < 8; r += 2) {
                v2f pr;
                pr.x = acc[nj][r];
                pr.y = acc[nj][r + 1];
                *(v2f*)(Dsm + swz(row0 + rst + r, col)) = pr;  // ds_store_b64
            }
        }
    }
}

extern "C" __global__ __launch_bounds__(256)
void ln_fused(const float* __restrict__ x,
              const float* __restrict__ Wlin,
              const float* __restrict__ Wrelu,
              const float* __restrict__ Wpool,
              float* __restrict__ ws_val,
              float* __restrict__ ws_vec)
{
    extern __shared__ float smem[];
    float* bufA = smem;          // swizzled panel: x1 -> x2 (in place)
    float* bufB = smem + PANEL;  // swizzled: x_in (128 rows) -> d -> d2

    const int tid  = threadIdx.x;
    const int lane = tid & 31;
    const int wave = tid >> 5;
    const int tile = blockIdx.x;      // 0..NT-1
    const int b    = blockIdx.y;      // 0..BATCH-1
    const int n0   = tile * TILE_N;

    // ---- stage x tile directly into swizzled layout:
    // thread gathers rows 4cg..4cg+3 at one column (4 coalesced b32 loads),
    // writes one contiguous b128 LDS store at swz(4cg, col).
    {
        const float* xb = x + ((size_t)b * CIN * 8) * NDIM + n0;
        for (int it = tid; it < (CIN / 4) * MCOLS; it += 256) {
            const int col = it & (MCOLS - 1);
            const int cg  = it >> 7;          // 4-row group index
            const int k8  = col >> 4;
            const int nn  = col & 15;
            const float* gp = xb + ((size_t)(cg << 2) * 8 + k8) * NDIM + nn;
            v4f v;
            v.x = gp[0];
            v.y = gp[8 * NDIM];
            v.z = gp[16 * NDIM];
            v.w = gp[24 * NDIM];
            *(v4f*)(bufB + (cg << 9) + (col << 2)) = v;
        }
    }
    __syncthreads();

    gemm_block(Wlin, CIN, CIN, bufB, bufA, wave, lane);    // x1 = W_lin @ x
    __syncthreads();
    gemm_block(Wrelu, COUT, COUT, bufA, bufB, wave, lane); // d = W_relu @ x1
    __syncthreads();

    // ---- Killing nonlinearity: x2 = (kf<0) ? x1 : x1 + kf*d   (in place in bufA)
    for (int pp = tid; pp < COUT * TILE_N; pp += 256) {
        const int f = pp >> 4;
        const int n = pp & 15;
        float xv[8], dv[8];
        #pragma unroll
        for (int k = 0; k < 8; ++k) {
            const int a = swz(f, k * TILE_N + n);
            xv[k] = bufA[a];
            dv[k] = bufB[a];
        }
        const float kf = killing6(xv, dv);
        #pragma unroll
        for (int k = 0; k < 8; ++k)
            bufA[swz(f, k * TILE_N + n)] = (kf < 0.0f) ? xv[k] : fmaf(kf, dv[k], xv[k]);
    }
    __syncthreads();

    gemm_block(Wpool, COUT, COUT, bufA, bufB, wave, lane); // d2 = W_pool @ x2
    __syncthreads();

    // ---- per-tile argmax of kf2 over n (strict > ascending => first occurrence)
    {
        const int f = tid;   // 256 threads <-> 256 channels
        float xv[8], dv[8];
        #pragma unroll
        for (int k = 0; k < 8; ++k) {
            const int a = swz(f, k * TILE_N);
            xv[k] = bufA[a];
            dv[k] = bufB[a];
        }
        float best = killing6(xv, dv);
        int bestn = 0;
        for (int n = 1; n < TILE_N; ++n) {
            #pragma unroll
            for (int k = 0; k < 8; ++k) {
                const int a = swz(f, k * TILE_N + n);
                xv[k] = bufA[a];
                dv[k] = bufB[a];
            }
            const float v = killing6(xv, dv);
            if (v > best) { best = v; bestn = n; }
        }
        const size_t base = ((size_t)b * COUT + f) * NT + tile;
        ws_val[base] = best;
        #pragma unroll
        for (int k = 0; k < 8; ++k)
            ws_vec[base * 8 + k] = bufA[swz(f, k * TILE_N + bestn)];
    }
}

extern "C" __global__ __launch_bounds__(256)
void ln_reduce(const float* __restrict__ ws_val,
               const float* __restrict__ ws_vec,
               float* __restrict__ out)
{
    const int bf = blockIdx.x * 256 + threadIdx.x;   // 0..BATCH*COUT-1
    if (bf >= BATCH * COUT) return;
    const float* v = ws_val + (size_t)bf * NT;
    float best = v[0];
    int bestt = 0;
    for (int t = 1; t < NT; ++t) {
        const float val = v[t];
        if (val > best) { best = val; bestt = t; }   // strict > keeps earliest tile
    }
    const float* src = ws_vec + ((size_t)bf * NT + bestt) * 8;
    #pragma unroll
    for (int k = 0; k < 8; ++k)
        out[(size_t)bf * 8 + k] = src[k];
}

extern "C" void kernel_launch(void* const* d_in, const int* in_sizes, int n_in,
                              void* d_out, int out_size, void* d_ws, size_t ws_size,
                              hipStream_t stream)
{
    (void)in_sizes; (void)n_in; (void)out_size; (void)ws_size;

    const float* x     = (const float*)d_in[0];   // [8,128,8,4096] f32
    const float* Wlin  = (const float*)d_in[1];   // [256,128] f32
    const float* Wrelu = (const float*)d_in[2];   // [256,256] f32
    const float* Wpool = (const float*)d_in[3];   // [256,256] f32
    float* out    = (float*)d_out;                // [8,256,8] f32
    float* ws_val = (float*)d_ws;                                   // [8*256*256]
    float* ws_vec = ws_val + (size_t)BATCH * COUT * NT;             // [8*256*256*8]

    const size_t shmem = (size_t)2 * PANEL * sizeof(float);         // 262144 B < 320 KB
    hipFuncSetAttribute((const void*)ln_fused,
                        hipFuncAttributeMaxDynamicSharedMemorySize, (int)shmem);

    dim3 grid(NT, BATCH);
    ln_fused<<<grid, 256, shmem, stream>>>(x, Wlin, Wrelu, Wpool, ws_val, ws_vec);
    ln_reduce<<<(BATCH * COUT + 255) / 256, 256, 0, stream>>>(ws_val, ws_vec, out);
}